// GFCND_12524124635536
// MI455X (gfx1250) — compile-verified
//
#include <hip/hip_runtime.h>
#include <math.h>

#define NNODES 8192
#define NEDGE  65536
#define NSAMP  4096
#define OUTC   32
#define EPS_BN 1e-5f

typedef float v2f __attribute__((ext_vector_type(2)));
typedef float v8f __attribute__((ext_vector_type(8)));

// ---------------- generic zero ----------------
__global__ void zero_k(float* __restrict__ p, int n) {
    int i = blockIdx.x * blockDim.x + threadIdx.x;
    if (i < n) p[i] = 0.0f;
}

// ---------------- SplineConv scatter ----------------
// one lane per (edge, channel): 32 lanes of a wave share one edge
__global__ void spline_scatter(const float* __restrict__ x, const int* __restrict__ ei,
                               const float* __restrict__ ea, const float* __restrict__ Wsp,
                               float* __restrict__ agg, float* __restrict__ cnt) {
    int gid = blockIdx.x * blockDim.x + threadIdx.x;
    int e = gid >> 5;
    int c = gid & 31;
    if (e >= NEDGE) return;
    int src = ei[e];
    int dst = ei[NEDGE + e];
    float v0 = ea[2 * e]     * 4.0f;   // (KS-1)=4
    float v1 = ea[2 * e + 1] * 4.0f;
    float f0 = floorf(v0), f1 = floorf(v1);
    float r0 = v0 - f0, r1 = v1 - f1;
    int i0 = (int)f0, i1 = (int)f1;
    float acc = 0.0f;
#pragma unroll
    for (int s = 0; s < 4; ++s) {
        int b0 = s & 1, b1 = (s >> 1) & 1;
        float bs = (b0 ? r0 : 1.0f - r0) * (b1 ? r1 : 1.0f - r1);
        int w0 = i0 + b0; w0 = w0 < 0 ? 0 : (w0 > 4 ? 4 : w0);
        int w1 = i1 + b1; w1 = w1 < 0 ? 0 : (w1 > 4 ? 4 : w1);
        acc += bs * Wsp[(w0 + w1 * 5) * OUTC + c];
    }
    atomicAdd(&agg[dst * OUTC + c], acc * x[src]);
    if (c == 0) atomicAdd(&cnt[dst], 1.0f);
}

// ---------------- SplineConv finalize: mean agg + root + bias + ELU ----------------
__global__ void spline_finalize(const float* __restrict__ x, const float* __restrict__ Wroot,
                                const float* __restrict__ bconv, const float* __restrict__ agg,
                                const float* __restrict__ cnt, float* __restrict__ h) {
    int gid = blockIdx.x * blockDim.x + threadIdx.x;
    int i = gid >> 5;
    int c = gid & 31;
    if (i >= NNODES) return;
    float a = agg[i * OUTC + c] / fmaxf(cnt[i], 1.0f);
    float t = a + x[i] * Wroot[c] + bconv[c];
    h[i * OUTC + c] = (t > 0.0f) ? t : expm1f(t);   // ELU alpha=1
}

// ---------------- Farthest point sampling: single persistent workgroup ----------------
// 1024 threads, 8 nodes/thread held in registers; per-step wave32 shfl reduce + LDS hop.
__global__ __launch_bounds__(1024) void fps_kernel(const float* __restrict__ pos,
                                                   int* __restrict__ idx) {
    const int tid  = threadIdx.x;
    const int lane = tid & 31;
    const int wid  = tid >> 5;
    float px[8], py[8], md[8];
#pragma unroll
    for (int k = 0; k < 8; ++k) {
        int j = tid + (k << 10);
        px[k] = pos[2 * j];
        py[k] = pos[2 * j + 1];
        md[k] = 1e10f;
    }
    __shared__ float s_v[32];
    __shared__ int   s_i[32];
    __shared__ float s_cur[2];
    if (tid == 0) { idx[0] = 0; s_cur[0] = pos[0]; s_cur[1] = pos[1]; }
    __syncthreads();
    float cx = s_cur[0], cy = s_cur[1];
    for (int it = 1; it < NSAMP; ++it) {
        float bv = -1.0f; int bi = 0x7fffffff;
#pragma unroll
        for (int k = 0; k < 8; ++k) {
            float dx = px[k] - cx, dy = py[k] - cy;
            float d = dx * dx + dy * dy;
            md[k] = fminf(md[k], d);
            if (md[k] > bv) { bv = md[k]; bi = tid + (k << 10); }  // earliest max kept
        }
#pragma unroll
        for (int off = 16; off > 0; off >>= 1) {            // wave32 argmax (first-max ties)
            float ov = __shfl_xor(bv, off, 32);
            int   oi = __shfl_xor(bi, off, 32);
            if (ov > bv || (ov == bv && oi < bi)) { bv = ov; bi = oi; }
        }
        if (lane == 0) { s_v[wid] = bv; s_i[wid] = bi; }
        __syncthreads();
        if (wid == 0) {
            bv = s_v[lane]; bi = s_i[lane];
#pragma unroll
            for (int off = 16; off > 0; off >>= 1) {
                float ov = __shfl_xor(bv, off, 32);
                int   oi = __shfl_xor(bi, off, 32);
                if (ov > bv || (ov == bv && oi < bi)) { bv = ov; bi = oi; }
            }
            if (lane == 0) {
                idx[it] = bi;
                s_cur[0] = pos[2 * bi];
                s_cur[1] = pos[2 * bi + 1];
            }
        }
        __syncthreads();
        cx = s_cur[0]; cy = s_cur[1];
    }
}

// ---------------- gather sampled features / positions ----------------
__global__ void gather_k(const float* __restrict__ h, const float* __restrict__ pos,
                         const int* __restrict__ idx, float* __restrict__ x_d,
                         float* __restrict__ pos_d) {
    int gid = blockIdx.x * blockDim.x + threadIdx.x;
    int j = gid >> 5;
    int c = gid & 31;
    if (j >= NSAMP) return;
    int s = idx[j];
    x_d[j * OUTC + c] = h[s * OUTC + c];
    if (c < 2) pos_d[j * 2 + c] = pos[s * 2 + c];
}

// ---------------- knn_interpolate (k=3) + skip-concat into padded feat[8192,36] ----------------
__global__ __launch_bounds__(256) void knn_feat(const float* __restrict__ pos,
                                                const float* __restrict__ x,
                                                const float* __restrict__ x_d,
                                                const float* __restrict__ pos_d,
                                                float* __restrict__ feat) {
    __shared__ float spd[NSAMP * 2];                       // 32 KB: all sampled positions
    for (int t = threadIdx.x; t < NSAMP * 2; t += 256) spd[t] = pos_d[t];
    __syncthreads();
    int i = blockIdx.x * 256 + threadIdx.x;
    if (i >= NNODES) return;
    float pix = pos[2 * i], piy = pos[2 * i + 1];
    float d0 = 1e30f, d1 = 1e30f, d2 = 1e30f;
    int   n0 = 0, n1 = 0, n2 = 0;
    for (int j = 0; j < NSAMP; ++j) {
        float dx = pix - spd[2 * j], dy = piy - spd[2 * j + 1];
        float d = dx * dx + dy * dy;
        if (d < d2) {                                      // strict <: earlier index wins ties
            if (d < d1) {
                if (d < d0) { d2 = d1; n2 = n1; d1 = d0; n1 = n0; d0 = d; n0 = j; }
                else        { d2 = d1; n2 = n1; d1 = d;  n1 = j; }
            } else          { d2 = d;  n2 = j; }
        }
    }
    float w0 = 1.0f / fmaxf(d0, 1e-16f);
    float w1 = 1.0f / fmaxf(d1, 1e-16f);
    float w2 = 1.0f / fmaxf(d2, 1e-16f);
    float inv = 1.0f / (w0 + w1 + w2);
    const float* r0 = x_d + n0 * OUTC;
    const float* r1 = x_d + n1 * OUTC;
    const float* r2 = x_d + n2 * OUTC;
    float* fo = feat + i * 36;
#pragma unroll
    for (int c = 0; c < OUTC; ++c)
        fo[c] = (r0[c] * w0 + r1[c] * w1 + r2[c] * w2) * inv;
    fo[32] = x[i];                                         // skip feature
    fo[33] = 0.0f; fo[34] = 0.0f; fo[35] = 0.0f;           // K-pad for WMMA
}

// ---------------- pad W1 [33,32] -> [36,32] ----------------
__global__ void padw1(const float* __restrict__ W1, float* __restrict__ W1p) {
    int gid = blockIdx.x * blockDim.x + threadIdx.x;
    if (gid >= 36 * 32) return;
    int r = gid >> 5, c = gid & 31;
    W1p[gid] = (r < 33) ? W1[r * 32 + c] : 0.0f;
}

// ---------------- fp32 WMMA GEMM: T = relu(A[8192,KPAD] * B[KPAD,32] + bias) ----------------
// one wave -> one 16x16 output tile; 1024 waves total; EXEC all-ones (exact tiling).
template <int KPAD>
__global__ __launch_bounds__(256) void mlp_wmma(const float* __restrict__ A, int lda,
                                                const float* __restrict__ B,
                                                const float* __restrict__ bias,
                                                float* __restrict__ T) {
    const int lane = threadIdx.x & 31;
    const int wid  = threadIdx.x >> 5;
    const int wg   = blockIdx.x * 8 + wid;       // 0..1023
    const int m0   = (wg >> 1) << 4;             // 512 M-tiles
    const int n0   = (wg & 1) << 4;              // 2  N-tiles
    const int half = lane >> 4;                  // lane half: K-pair select
    const int l15  = lane & 15;
    const float* arow = A + (m0 + l15) * lda;
    v8f acc = {};
#pragma unroll
    for (int k0 = 0; k0 < KPAD; k0 += 4) {
        int kb = k0 + (half << 1);               // lanes 0-15: K=k0,k0+1 ; 16-31: K=k0+2,k0+3
        v2f a, b;
        a.x = arow[kb];
        a.y = arow[kb + 1];
        b.x = B[kb * 32 + n0 + l15];
        b.y = B[(kb + 1) * 32 + n0 + l15];
        acc = __builtin_amdgcn_wmma_f32_16x16x4_f32(false, a, false, b,
                                                    (short)0, acc, false, false);
    }
    float bia = bias[n0 + l15];
#pragma unroll
    for (int r = 0; r < 8; ++r) {                // D VGPR r -> rows m0+r (lanes 0-15), m0+r+8
        float v = acc[r] + bia;
        v = fmaxf(v, 0.0f);                      // ReLU
        T[(m0 + r + (half << 3)) * 32 + n0 + l15] = v;
    }
}

// ---------------- BatchNorm: per-column batch stats (biased var) ----------------
__global__ __launch_bounds__(256) void bn_stats(const float* __restrict__ T,
                                                float* __restrict__ mv) {
    int c = blockIdx.x;                          // 32 blocks, one per column
    float s = 0.0f, ss = 0.0f;
    for (int i = threadIdx.x; i < NNODES; i += 256) {
        float v = T[i * 32 + c];
        s += v; ss += v * v;
    }
    __shared__ float sh[256], sh2[256];
    sh[threadIdx.x] = s; sh2[threadIdx.x] = ss;
    __syncthreads();
    for (int o = 128; o > 0; o >>= 1) {
        if (threadIdx.x < o) { sh[threadIdx.x] += sh[threadIdx.x + o];
                               sh2[threadIdx.x] += sh2[threadIdx.x + o]; }
        __syncthreads();
    }
    if (threadIdx.x == 0) {
        float mu = sh[0] / (float)NNODES;
        mv[c]      = mu;
        mv[32 + c] = sh2[0] / (float)NNODES - mu * mu;
    }
}

__global__ void bn_apply(const float* __restrict__ T, const float* __restrict__ mv,
                         const float* __restrict__ g, const float* __restrict__ be,
                         float* __restrict__ H) {
    int gid = blockIdx.x * blockDim.x + threadIdx.x;
    int i = gid >> 5, c = gid & 31;
    if (i >= NNODES) return;
    H[i * 32 + c] = g[c] * (T[i * 32 + c] - mv[c]) * rsqrtf(mv[32 + c] + EPS_BN) + be[c];
}

// ---------------- layer3: [8192,32] x [32,1] + relu ----------------
__global__ void layer3(const float* __restrict__ H, const float* __restrict__ W3,
                       const float* __restrict__ b3, float* __restrict__ t3) {
    int i = blockIdx.x * blockDim.x + threadIdx.x;
    if (i >= NNODES) return;
    float s = b3[0];
#pragma unroll
    for (int c = 0; c < 32; ++c) s += H[i * 32 + c] * W3[c];
    t3[i] = fmaxf(s, 0.0f);
}

__global__ __launch_bounds__(1024) void stats3(const float* __restrict__ t3,
                                               float* __restrict__ mv) {
    float s = 0.0f, ss = 0.0f;
    for (int i = threadIdx.x; i < NNODES; i += 1024) { float v = t3[i]; s += v; ss += v * v; }
    __shared__ float sh[1024], sh2[1024];
    sh[threadIdx.x] = s; sh2[threadIdx.x] = ss;
    __syncthreads();
    for (int o = 512; o > 0; o >>= 1) {
        if (threadIdx.x < o) { sh[threadIdx.x] += sh[threadIdx.x + o];
                               sh2[threadIdx.x] += sh2[threadIdx.x + o]; }
        __syncthreads();
    }
    if (threadIdx.x == 0) {
        float mu = sh[0] / (float)NNODES;
        mv[0] = mu;
        mv[1] = sh2[0] / (float)NNODES - mu * mu;
    }
}

__global__ void final_k(const float* __restrict__ t3, const float* __restrict__ mv,
                        const float* __restrict__ g3, const float* __restrict__ be3,
                        float* __restrict__ out) {
    int i = blockIdx.x * blockDim.x + threadIdx.x;
    if (i >= NNODES) return;
    float z = g3[0] * (t3[i] - mv[0]) * rsqrtf(mv[1] + EPS_BN) + be3[0];
    out[i] = 1.0f / (1.0f + expf(-z));
}

// ---------------- launcher ----------------
extern "C" void kernel_launch(void* const* d_in, const int* in_sizes, int n_in,
                              void* d_out, int out_size, void* d_ws, size_t ws_size,
                              hipStream_t stream) {
    (void)in_sizes; (void)n_in; (void)out_size; (void)ws_size;
    const float* x     = (const float*)d_in[0];
    const float* pos   = (const float*)d_in[1];
    const int*   ei    = (const int*)d_in[2];
    const float* ea    = (const float*)d_in[3];
    const float* Wsp   = (const float*)d_in[4];
    const float* Wroot = (const float*)d_in[5];
    const float* bconv = (const float*)d_in[6];
    const float* W1  = (const float*)d_in[7];
    const float* b1  = (const float*)d_in[8];
    const float* g1  = (const float*)d_in[9];
    const float* be1 = (const float*)d_in[10];
    const float* W2  = (const float*)d_in[11];
    const float* b2  = (const float*)d_in[12];
    const float* g2  = (const float*)d_in[13];
    const float* be2 = (const float*)d_in[14];
    const float* W3  = (const float*)d_in[15];
    const float* b3  = (const float*)d_in[16];
    const float* g3  = (const float*)d_in[17];
    const float* be3 = (const float*)d_in[18];
    float* out = (float*)d_out;

    char* ws = (char*)d_ws;
    size_t off = 0;
    auto alloc = [&](size_t bytes) -> char* {
        char* p = ws + off;
        off += (bytes + 255) & ~(size_t)255;
        return p;
    };
    float* agg   = (float*)alloc((size_t)NNODES * 33 * 4);   // agg[8192*32] + cnt[8192]
    float* cnt   = agg + (size_t)NNODES * OUTC;
    float* h     = (float*)alloc((size_t)NNODES * OUTC * 4);
    int*   idx   = (int*)  alloc((size_t)NSAMP * 4);
    float* pos_d = (float*)alloc((size_t)NSAMP * 2 * 4);
    float* x_d   = (float*)alloc((size_t)NSAMP * OUTC * 4);
    float* feat  = (float*)alloc((size_t)NNODES * 36 * 4);
    float* W1p   = (float*)alloc((size_t)36 * 32 * 4);
    float* t1    = (float*)alloc((size_t)NNODES * 32 * 4);
    float* h1    = (float*)alloc((size_t)NNODES * 32 * 4);
    float* t2    = (float*)alloc((size_t)NNODES * 32 * 4);
    float* h2    = (float*)alloc((size_t)NNODES * 32 * 4);
    float* t3    = (float*)alloc((size_t)NNODES * 4);
    float* mv1   = (float*)alloc(64 * 4);
    float* mv2   = (float*)alloc(64 * 4);
    float* mv3   = (float*)alloc(2 * 4);

    // SplineConv
    zero_k<<<(NNODES * 33 + 255) / 256, 256, 0, stream>>>(agg, NNODES * 33);
    spline_scatter<<<(NEDGE * 32) / 256, 256, 0, stream>>>(x, ei, ea, Wsp, agg, cnt);
    spline_finalize<<<(NNODES * 32) / 256, 256, 0, stream>>>(x, Wroot, bconv, agg, cnt, h);
    // FPS (sequential, single persistent workgroup)
    fps_kernel<<<1, 1024, 0, stream>>>(pos, idx);
    gather_k<<<(NSAMP * 32) / 256, 256, 0, stream>>>(h, pos, idx, x_d, pos_d);
    // kNN-interpolate + concat
    knn_feat<<<NNODES / 256, 256, 0, stream>>>(pos, x, x_d, pos_d, feat);
    // MLP (fp32 WMMA) + BN
    padw1<<<(36 * 32 + 255) / 256, 256, 0, stream>>>(W1, W1p);
    mlp_wmma<36><<<128, 256, 0, stream>>>(feat, 36, W1p, b1, t1);
    bn_stats<<<32, 256, 0, stream>>>(t1, mv1);
    bn_apply<<<(NNODES * 32) / 256, 256, 0, stream>>>(t1, mv1, g1, be1, h1);
    mlp_wmma<32><<<128, 256, 0, stream>>>(h1, 32, W2, b2, t2);
    bn_stats<<<32, 256, 0, stream>>>(t2, mv2);
    bn_apply<<<(NNODES * 32) / 256, 256, 0, stream>>>(t2, mv2, g2, be2, h2);
    layer3<<<NNODES / 256, 256, 0, stream>>>(h2, W3, b3, t3);
    stats3<<<1, 1024, 0, stream>>>(t3, mv3);
    final_k<<<NNODES / 256, 256, 0, stream>>>(t3, mv3, g3, be3, out);
}